// MonarchMixerLayer_8761733283948
// MI455X (gfx1250) — compile-verified
//
#include <hip/hip_runtime.h>
#include <stdint.h>

// Monarch mixer fused kernel for gfx1250 (MI455X), full f32 precision via
// V_WMMA_F32_16X16X4_F32, with double-buffered async X staging
// (GLOBAL_LOAD_ASYNC_TO_LDS_B128 + s_wait_asynccnt).
//
// Shapes: x[4,4096,1024] f32, L/R [4,256,256] f32, perm[1024] i64.
// Each workgroup: ROWS=32 rows x full H=1024 features, fused GEMM1+GEMM2+perm.

typedef __attribute__((ext_vector_type(2))) float v2f;
typedef __attribute__((ext_vector_type(8))) float v8f;

#define ROWS     32      // rows per workgroup
#define BS       256     // block size (H / nblocks)
#define HFEAT    1024    // hidden dim
#define NBLK     4
#define XP       260     // Xs LDS pitch (pad -> conflict-free b64 A loads)
#define YP       260     // Y1s LDS pitch
#define OPITCH   1040    // output row-buffer pitch (pad 16 -> disjoint bank halves)
#define NTHREADS 512     // 16 wave32 waves

// Issue this thread's share (4 x B128) of the async copy of X block-column
// `n` into LDS buffer `xs`. Tracked on ASYNCcnt.
__device__ __forceinline__ void stage_x_async(const float* __restrict__ x,
                                              int row_base, int n,
                                              float* xs, int tid)
{
    const float* Xsrc = x + (size_t)row_base * HFEAT + n * BS;
    #pragma unroll
    for (int i = 0; i < 4; ++i) {
        int idx = tid + i * NTHREADS;          // 0 .. 2047 (float4 units)
        int r   = idx >> 6;                    // 64 float4 per row
        int c4  = (idx & 63) << 2;
        unsigned long long gaddr =
            (unsigned long long)(uintptr_t)(Xsrc + (size_t)r * HFEAT + c4);
        unsigned lds_off = (unsigned)(uintptr_t)(&xs[r * XP + c4]); // addr[31:0]
        asm volatile("global_load_async_to_lds_b128 %0, %1, off"
                     :: "v"(lds_off), "v"(gaddr) : "memory");
    }
}

__global__ __launch_bounds__(NTHREADS) void monarch_mixer_kernel(
    const float* __restrict__ x,
    const float* __restrict__ Lb,
    const float* __restrict__ Rb,
    const long long* __restrict__ perm,
    float* __restrict__ out)
{
    extern __shared__ float smem[];
    float* Xs0  = smem;                        // ROWS * XP floats (ping)
    float* Xs1  = Xs0  + ROWS * XP;            // ROWS * XP floats (pong)
    float* Y1s  = Xs1  + ROWS * XP;            // ROWS * YP floats
    float* Outs = Y1s  + ROWS * YP;            // ROWS * OPITCH floats
    int*   Ps   = (int*)(Outs + ROWS * OPITCH); // HFEAT ints

    const int tid   = threadIdx.x;
    const int lane  = tid & 31;
    const int wv    = tid >> 5;        // 0..15 : N-strip owner
    const int lrow  = lane & 15;       // M / N index within fragment
    const int khalf = lane >> 4;       // 0: K pair {0,1}; 1: K pair {2,3}
    const int row_base = blockIdx.x * ROWS;
    const int ncol  = wv * 16 + lrow;  // column within a 256-wide block

    // Preload permutation (values < 1024 fit in int)
    for (int h = tid; h < HFEAT; h += NTHREADS) Ps[h] = (int)perm[h];

    // Prologue: start async stage of block 0 into Xs0.
    stage_x_async(x, row_base, 0, Xs0, tid);

    for (int n = 0; n < NBLK; ++n) {
        float* Xcur = (n & 1) ? Xs1 : Xs0;

        if (n < NBLK - 1) {
            // Overlap: stream block n+1 while we compute block n.
            // Target buffer was last read in block n-1, released by the
            // end-of-iteration barrier.
            stage_x_async(x, row_base, n + 1, (n & 1) ? Xs0 : Xs1, tid);
            // 4 newly-issued outstanding => block n's 4 transfers retired.
            asm volatile("s_wait_asynccnt 0x4" ::: "memory");
        } else {
            asm volatile("s_wait_asynccnt 0x0" ::: "memory");
        }
        __syncthreads();   // make every wave's staged data visible to all

        const float* Ln = Lb + (size_t)n * BS * BS;
        const float* Rn = Rb + (size_t)n * BS * BS;

        // ---- GEMM1: Y1 = Xblk @ L[n] ------------------------------------
        v8f acc0 = {}; v8f acc1 = {};
        #pragma unroll 4
        for (int k0 = 0; k0 < BS; k0 += 4) {
            const int kk = k0 + 2 * khalf;
            v2f bf;
            bf.x = Ln[(size_t)kk * BS + ncol];
            bf.y = Ln[(size_t)(kk + 1) * BS + ncol];
            v2f af0 = *(const v2f*)(&Xcur[lrow * XP + kk]);
            v2f af1 = *(const v2f*)(&Xcur[(16 + lrow) * XP + kk]);
            acc0 = __builtin_amdgcn_wmma_f32_16x16x4_f32(
                false, af0, false, bf, (short)0, acc0, false, false);
            acc1 = __builtin_amdgcn_wmma_f32_16x16x4_f32(
                false, af1, false, bf, (short)0, acc1, false, false);
        }
        #pragma unroll
        for (int r = 0; r < 8; ++r) {
            Y1s[(r + 8 * khalf) * YP + ncol]      = acc0[r];
            Y1s[(16 + r + 8 * khalf) * YP + ncol] = acc1[r];
        }
        __syncthreads();

        // ---- GEMM2: Y2 = Y1 @ R[n], into row buffer at column slot n ----
        v8f c0 = {}; v8f c1 = {};
        #pragma unroll 4
        for (int k0 = 0; k0 < BS; k0 += 4) {
            const int kk = k0 + 2 * khalf;
            v2f bf;
            bf.x = Rn[(size_t)kk * BS + ncol];
            bf.y = Rn[(size_t)(kk + 1) * BS + ncol];
            v2f af0 = *(const v2f*)(&Y1s[lrow * YP + kk]);
            v2f af1 = *(const v2f*)(&Y1s[(16 + lrow) * YP + kk]);
            c0 = __builtin_amdgcn_wmma_f32_16x16x4_f32(
                false, af0, false, bf, (short)0, c0, false, false);
            c1 = __builtin_amdgcn_wmma_f32_16x16x4_f32(
                false, af1, false, bf, (short)0, c1, false, false);
        }
        #pragma unroll
        for (int r = 0; r < 8; ++r) {
            Outs[(r + 8 * khalf) * OPITCH + n * BS + ncol]      = c0[r];
            Outs[(16 + r + 8 * khalf) * OPITCH + n * BS + ncol] = c1[r];
        }
        __syncthreads();   // releases Y1s + the Xs buffer read this iter
    }

    // ---- permutation gather + coalesced writeout ------------------------
    for (int idx = tid; idx < ROWS * HFEAT; idx += NTHREADS) {
        int r = idx >> 10;
        int h = idx & (HFEAT - 1);
        out[(size_t)(row_base + r) * HFEAT + h] = Outs[r * OPITCH + Ps[h]];
    }
}

extern "C" void kernel_launch(void* const* d_in, const int* in_sizes, int n_in,
                              void* d_out, int out_size, void* d_ws, size_t ws_size,
                              hipStream_t stream) {
    const float*     xin  = (const float*)d_in[0];
    const float*     Lb   = (const float*)d_in[1];
    const float*     Rb   = (const float*)d_in[2];
    const long long* perm = (const long long*)d_in[3];
    float*           outp = (float*)d_out;

    const int rows = in_sizes[0] / HFEAT;       // B*S = 16384
    const int grid = rows / ROWS;               // 512 workgroups

    const size_t lds_bytes =
        (size_t)(2 * ROWS * XP + ROWS * YP + ROWS * OPITCH) * sizeof(float)
        + (size_t)HFEAT * sizeof(int);          // ~237 KB, fits 320 KB WGP LDS

    monarch_mixer_kernel<<<dim3(grid), dim3(NTHREADS), lds_bytes, stream>>>(
        xin, Lb, Rb, perm, outp);
}